// BatchRankingLoss_10496900071731
// MI455X (gfx1250) — compile-verified
//
#include <hip/hip_runtime.h>

#define MARGIN 2.0f
#define TILE 128          // 128 threads/block = 4 wave32 waves; 64x64 tile grid for n=8192

typedef __attribute__((ext_vector_type(2))) float v2f;
typedef __attribute__((ext_vector_type(8))) float v8f;

// Block (ci, cj) with ci <= cj computes the hinge sum over the TILE x TILE tile
// of pairs (i in chunk ci, j in chunk cj), with i < j enforced on the diagonal.
__global__ __launch_bounds__(TILE)
void rank_tile_kernel(const float* __restrict__ preds,
                      const float* __restrict__ labels,
                      float* __restrict__ partials,
                      int n, int nchunks) {
  const int cj  = blockIdx.x;
  const int ci  = blockIdx.y;
  const int tid = threadIdx.x;
  const int pidx = cj * nchunks + ci;

  __shared__ float sp[TILE];
  __shared__ float sl[TILE];
  __shared__ float wsum[TILE / 32];

  if (ci > cj) {                      // lower-triangle tile: contributes nothing
    if (tid == 0) partials[pidx] = 0.0f;
    return;
  }

  const int   j  = cj * TILE + tid;
  const bool  jv = (j < n);
  const float pj = jv ? preds[j]  : 0.0f;
  const float lj = jv ? labels[j] : 0.0f;

  const int gi = ci * TILE + tid;
  sp[tid] = (gi < n) ? preds[gi]  : 0.0f;
  sl[tid] = (gi < n) ? labels[gi] : 0.0f;
  __syncthreads();

  const int limit = min(TILE, n - ci * TILE);
  float acc = 0.0f;

  if (ci < cj) {
    // fully off-diagonal: every i in this chunk satisfies i < j
    #pragma unroll 4
    for (int k = 0; k < limit; ++k) {
      float pd = pj - sp[k];                 // p_j - p_i
      float dl = lj - sl[k];                 // l_j - l_i
      float s  = (dl > 0.0f) ? 1.0f : ((dl < 0.0f) ? -1.0f : 0.0f);
      float t  = __builtin_fmaf(-pd, s, MARGIN);
      acc += __builtin_fmaxf(t, 0.0f);
    }
  } else {
    // diagonal tile: need i < j  <=>  k < tid (same chunk)
    #pragma unroll 4
    for (int k = 0; k < limit; ++k) {
      float pd = pj - sp[k];
      float dl = lj - sl[k];
      float s  = (dl > 0.0f) ? 1.0f : ((dl < 0.0f) ? -1.0f : 0.0f);
      float t  = __builtin_fmaf(-pd, s, MARGIN);
      float h  = __builtin_fmaxf(t, 0.0f);
      acc += (k < tid) ? h : 0.0f;
    }
  }
  if (!jv) acc = 0.0f;

  // ---- wave-level reduction on the matrix pipe: D = A(16x4) * ones(4x16) ----
  // A layout (32-bit A, 16x4): VGPR0 lanes 0-15 -> K=0 (M=lane),
  //                            VGPR0 lanes 16-31 -> K=2 (M=lane-16); VGPR1 -> K=1/K=3.
  // With a = {acc, 0}: rowsum[m] = acc[m] + acc[m+16].
  // D layout: lane L<16 VGPR r holds D[r][L]; lane L>=16 holds D[8+r][L-16].
  v2f a; a[0] = acc;  a[1] = 0.0f;
  v2f b; b[0] = 1.0f; b[1] = 1.0f;       // ones matrix
  v8f c = {};
  c = __builtin_amdgcn_wmma_f32_16x16x4_f32(
      /*neg_a=*/false, a, /*neg_b=*/false, b,
      /*c_mod=*/(short)0, c, /*reuse_a=*/false, /*reuse_b=*/false);
  float s8 = ((c[0] + c[1]) + (c[2] + c[3])) + ((c[4] + c[5]) + (c[6] + c[7]));
  // lanes 0-15 now hold sum(rowsum[0..7]); lanes 16-31 hold sum(rowsum[8..15])
  float wavesum = s8 + __shfl_xor(s8, 16, 32);   // identical on all 32 lanes

  const int wave = tid >> 5;
  const int lane = tid & 31;
  if (lane == 0) wsum[wave] = wavesum;
  __syncthreads();
  if (tid == 0) {
    float bsum = 0.0f;
    #pragma unroll
    for (int w = 0; w < TILE / 32; ++w) bsum += wsum[w];
    partials[pidx] = bsum;
  }
}

// Deterministic final reduction: fixed strided assignment + fixed LDS tree.
__global__ __launch_bounds__(256)
void reduce_kernel(const float* __restrict__ partials, float* __restrict__ out, int count) {
  __shared__ float sm[256];
  const int tid = threadIdx.x;
  float s = 0.0f;
  for (int i = tid; i < count; i += 256) s += partials[i];
  sm[tid] = s;
  __syncthreads();
  #pragma unroll
  for (int off = 128; off > 0; off >>= 1) {
    if (tid < off) sm[tid] += sm[tid + off];
    __syncthreads();
  }
  if (tid == 0) out[0] = sm[0];
}

extern "C" void kernel_launch(void* const* d_in, const int* in_sizes, int n_in,
                              void* d_out, int out_size, void* d_ws, size_t ws_size,
                              hipStream_t stream) {
  const float* preds  = (const float*)d_in[0];
  const float* labels = (const float*)d_in[1];
  float* out = (float*)d_out;
  const int n = in_sizes[0];
  const int nchunks = (n + TILE - 1) / TILE;

  float* partials = (float*)d_ws;              // nchunks^2 floats (16 KB for n=8192)

  dim3 grid(nchunks, nchunks);
  rank_tile_kernel<<<grid, TILE, 0, stream>>>(preds, labels, partials, n, nchunks);
  reduce_kernel<<<1, 256, 0, stream>>>(partials, out, nchunks * nchunks);
}